// BPR_85796266705487
// MI455X (gfx1250) — compile-verified
//
#include <hip/hip_runtime.h>

// Problem constants (from reference).
#define U_NUM 52643
#define I_NUM 91599
#define NROW  (U_NUM + I_NUM)     // 144242
#define FDIM  64
#define E_NUM 2000000

#define TPB         256           // 8 wave32s per block
#define EDGE_BLOCKS 4096          // grid-stride over E*16 work items
#define NVEC        (NROW * (FDIM / 4))   // float4 elements per matrix

// Hardware fp32 atomic add (GLOBAL_ATOMIC_ADD_F32, RMW executed at the L2
// atomic units — entire working set is L2-resident in MI455X's 192MB L2).
__device__ __forceinline__ void atom_add_f32(float* p, float v) {
#if defined(__HIP_PLATFORM_AMD__) || defined(__AMDGCN__)
    unsafeAtomicAdd(p, v);
#else
    atomicAdd(p, v);
#endif
}

// ---- Defined FIRST so the disasm snippet shows the atomic/prefetch path ----
// Adjacency propagation: for each edge e (16 lanes, one float4 per lane):
//   cur[user_row]  += adj_vals[e]     * embed_item[ui_i[e]]
//   cur[item_row]  += adj_vals[e + E] * embed_user[ui_u[e]]
__global__ void bpr_adj(const int*   __restrict__ uiu,
                        const int*   __restrict__ uii,
                        const float* __restrict__ adjv,
                        const float4* __restrict__ eu4,
                        const float4* __restrict__ ei4,
                        float* __restrict__ cur) {
    const int total  = E_NUM * 16;
    const int stride = gridDim.x * blockDim.x;     // multiple of 16
    for (int t = blockIdx.x * blockDim.x + threadIdx.x; t < total; t += stride) {
        int e = t >> 4;
        int f = t & 15;
        int ep = e + (stride >> 4);                // next sweep's edge
        if (ep < E_NUM) {                          // gfx1250 global_prefetch_b8
            __builtin_prefetch(uiu + ep, 0, 1);
            __builtin_prefetch(uii + ep, 0, 1);
            __builtin_prefetch(adjv + ep, 0, 1);
        }
        int   u    = uiu[e];
        int   it   = uii[e];
        float a_u  = adjv[e];                      // item -> user row
        float a_i  = adjv[e + E_NUM];              // user -> item row
        float4 vi  = ei4[it * 16 + f];             // 256B coalesced per 16 lanes
        float4 vu  = eu4[u  * 16 + f];
        float* du  = cur + (size_t)u * FDIM + f * 4;
        float* dt  = cur + ((size_t)U_NUM + it) * FDIM + f * 4;
        atom_add_f32(du + 0, a_u * vi.x);
        atom_add_f32(du + 1, a_u * vi.y);
        atom_add_f32(du + 2, a_u * vi.z);
        atom_add_f32(du + 3, a_u * vi.w);
        atom_add_f32(dt + 0, a_i * vu.x);
        atom_add_f32(dt + 1, a_i * vu.y);
        atom_add_f32(dt + 2, a_i * vu.z);
        atom_add_f32(dt + 3, a_i * vu.w);
    }
}

// One propagation step (reads only `cur`, scatters into `nxt`):
//   nxt_u[ui_u[e]] += ui_vals[e] * cur_i[ui_i[e]]
//   nxt_i[ui_i[e]] += iu_vals[e] * cur_u[ui_u[e]]
__global__ void bpr_spmm(const int*   __restrict__ uiu,
                         const int*   __restrict__ uii,
                         const float* __restrict__ uiv,
                         const float* __restrict__ iuv,
                         const float* __restrict__ cur,
                         float* __restrict__ nxt) {
    const int total  = E_NUM * 16;
    const int stride = gridDim.x * blockDim.x;
    const float4* cu4 = (const float4*)cur;
    const float4* ci4 = cu4 + (size_t)U_NUM * (FDIM / 4);
    float* nu = nxt;
    float* ni = nxt + (size_t)U_NUM * FDIM;
    for (int t = blockIdx.x * blockDim.x + threadIdx.x; t < total; t += stride) {
        int e = t >> 4;
        int f = t & 15;
        int ep = e + (stride >> 4);
        if (ep < E_NUM) {
            __builtin_prefetch(uiu + ep, 0, 1);
            __builtin_prefetch(uii + ep, 0, 1);
            __builtin_prefetch(uiv + ep, 0, 1);
            __builtin_prefetch(iuv + ep, 0, 1);
        }
        int   u  = uiu[e];
        int   it = uii[e];
        float wu = uiv[e];                         // item feat -> user row
        float wi = iuv[e];                         // user feat -> item row
        float4 xi = ci4[it * 16 + f];
        float4 xu = cu4[u  * 16 + f];
        float* pu = nu + (size_t)u  * FDIM + f * 4;
        float* pi = ni + (size_t)it * FDIM + f * 4;
        atom_add_f32(pu + 0, wu * xi.x);
        atom_add_f32(pu + 1, wu * xi.y);
        atom_add_f32(pu + 2, wu * xi.z);
        atom_add_f32(pu + 3, wu * xi.w);
        atom_add_f32(pi + 0, wi * xu.x);
        atom_add_f32(pi + 1, wi * xu.y);
        atom_add_f32(pi + 2, wi * xu.z);
        atom_add_f32(pi + 3, wi * xu.w);
    }
}

// cur = concat(embed_user, embed_item)   (float4 vectorized copy)
__global__ void bpr_init_cur(const float4* __restrict__ eu,
                             const float4* __restrict__ ei,
                             float4* __restrict__ cur) {
    int i = blockIdx.x * blockDim.x + threadIdx.x;
    const int nu = U_NUM * (FDIM / 4);
    if (i < NVEC) cur[i] = (i < nu) ? eu[i] : ei[i - nu];
}

// Fused: out (+)= w_prev * cur   AND   nxt = cur * diag(d_i ++ d_j).
// One read of `cur` feeds both the layer accumulation and the next layer's
// diagonal term (saves a full 37MB pass per iteration vs separate kernels).
__global__ void bpr_diag_axpy(float4* __restrict__ nxt,
                              const float4* __restrict__ cur,
                              const float* __restrict__ d_i,
                              const float* __restrict__ d_j,
                              float4* __restrict__ out,
                              float w, int init) {
    int i = blockIdx.x * blockDim.x + threadIdx.x;
    if (i >= NVEC) return;
    int r = i >> 4;
    float d = (r < U_NUM) ? d_i[r] : d_j[r - U_NUM];
    float4 c = cur[i];
    float4 o;
    if (init) {
        o.x = w * c.x; o.y = w * c.y; o.z = w * c.z; o.w = w * c.w;
    } else {
        o = out[i];
        o.x += w * c.x; o.y += w * c.y; o.z += w * c.z; o.w += w * c.w;
    }
    out[i] = o;
    float4 n;
    n.x = c.x * d; n.y = c.y * d; n.z = c.z * d; n.w = c.w * d;
    nxt[i] = n;
}

// Final accumulation only: out += w * src
__global__ void bpr_axpy(float4* __restrict__ out,
                         const float4* __restrict__ src,
                         float w) {
    int i = blockIdx.x * blockDim.x + threadIdx.x;
    if (i >= NVEC) return;
    float4 s = src[i];
    float4 o = out[i];
    o.x += w * s.x; o.y += w * s.y; o.z += w * s.z; o.w += w * s.w;
    out[i] = o;
}

extern "C" void kernel_launch(void* const* d_in, const int* in_sizes, int n_in,
                              void* d_out, int out_size, void* d_ws, size_t ws_size,
                              hipStream_t stream) {
    const float* eu   = (const float*)d_in[0];   // embed_user (U,64)
    const float* ei   = (const float*)d_in[1];   // embed_item (I,64)
    const float* di   = (const float*)d_in[2];   // d_i (U)
    const float* dj   = (const float*)d_in[3];   // d_j (I)
    const int*   uiu  = (const int*)  d_in[4];   // ui_u (E)
    const int*   uii  = (const int*)  d_in[5];   // ui_i (E)
    const float* uiv  = (const float*)d_in[6];   // ui_vals (E)
    const float* iuv  = (const float*)d_in[7];   // iu_vals (E)
    const float* adjv = (const float*)d_in[8];   // adj_vals (2E)

    float* out = (float*)d_out;                  // (U+I)*64: users then items
    float* cur = (float*)d_ws;                   // layer k   (NROW*64 f32)
    float* nxt = cur + (size_t)NROW * FDIM;      // layer k+1 (NROW*64 f32)

    const int gElem = (NVEC + TPB - 1) / TPB;

    // Layer 0: cur = ego + adjacency-propagated ego
    bpr_init_cur<<<gElem, TPB, 0, stream>>>((const float4*)eu, (const float4*)ei,
                                            (float4*)cur);
    bpr_adj<<<EDGE_BLOCKS, TPB, 0, stream>>>(uiu, uii, adjv,
                                             (const float4*)eu, (const float4*)ei,
                                             cur);

    // Layer weights w_k for k = 0..4: 1, 1/2, 1/3, 1/4, 1.
    // Iteration k accumulates out (+)= w[k-1]*layer(k-1) fused with the
    // diagonal-scale that seeds layer k, then scatters the SpMM term.
    const float w[5] = {1.0f, 0.5f, 1.0f / 3.0f, 0.25f, 1.0f};
    for (int k = 1; k <= 4; ++k) {
        bpr_diag_axpy<<<gElem, TPB, 0, stream>>>((float4*)nxt, (const float4*)cur,
                                                 di, dj, (float4*)out,
                                                 w[k - 1], (k == 1) ? 1 : 0);
        bpr_spmm<<<EDGE_BLOCKS, TPB, 0, stream>>>(uiu, uii, uiv, iuv, cur, nxt);
        float* tmp = cur; cur = nxt; nxt = tmp;  // deterministic ping-pong
    }
    bpr_axpy<<<gElem, TPB, 0, stream>>>((float4*)out, (const float4*)cur, w[4]);
}